// Encoder_eCOLGAT_42356967473563
// MI455X (gfx1250) — compile-verified
//
#include <hip/hip_runtime.h>
#include <hip/hip_bf16.h>

typedef __attribute__((ext_vector_type(2))) float v2f;
typedef __attribute__((ext_vector_type(8))) float v8f;

#define HEADS 4
#define NEG_SLOPE 0.2f

// ---------- monotonic float <-> uint key for atomicMax-based segment max ----------
__device__ __forceinline__ unsigned fkey(float f) {
  unsigned b = __float_as_uint(f);
  return (b & 0x80000000u) ? ~b : (b | 0x80000000u);
}
__device__ __forceinline__ float funkey(unsigned k) {
  unsigned b = (k & 0x80000000u) ? (k & 0x7FFFFFFFu) : ~k;
  return __uint_as_float(b);
}
#define NEG_INF_KEY 0x007FFFFFu   // fkey(-inf)

// =====================================================================
// GEMM: C[M x Nc] = A[M x K] @ B[K x Nc], row-major, fp32 WMMA 16x16x4.
// One wave computes a 16x64 output strip (4 accumulators), block = 8 waves.
// Requires M % 16 == 0, K % 4 == 0, Nc % 64 == 0 (true for all our shapes).
// A fragment (ISA 7.12.2, 32-bit A 16x4): lanes 0-15 hold K=k..k+1,
// lanes 16-31 hold K=k+2..k+3. C/D: VGPR r -> M = r + 8*(lane>=16), N = lane&15.
// =====================================================================
__global__ __launch_bounds__(256)
void gemm_f32_wmma(const float* __restrict__ A, const float* __restrict__ B,
                   float* __restrict__ C, int M, int K, int Nc) {
  const int lane  = threadIdx.x & 31;
  const int wave  = threadIdx.x >> 5;
  const int tileM = (blockIdx.y * 8 + wave) * 16;
  const int tileN = blockIdx.x * 64;
  if (tileM >= M) return;

  const int half = lane >> 4;   // 0 for lanes 0-15, 1 for lanes 16-31
  const int l16  = lane & 15;
  const int arow = tileM + l16;

  v8f acc0 = {}, acc1 = {}, acc2 = {}, acc3 = {};

  for (int k = 0; k < K; k += 4) {
    const int ka = k + 2 * half;            // this half-wave's K pair
    v2f a;
    a.x = A[(size_t)arow * K + ka];
    a.y = A[(size_t)arow * K + ka + 1];

    const float* __restrict__ Br0 = B + (size_t)ka * Nc + tileN + l16;
    const float* __restrict__ Br1 = B + (size_t)(ka + 1) * Nc + tileN + l16;
    v2f b0; b0.x = Br0[0];  b0.y = Br1[0];
    v2f b1; b1.x = Br0[16]; b1.y = Br1[16];
    v2f b2; b2.x = Br0[32]; b2.y = Br1[32];
    v2f b3; b3.x = Br0[48]; b3.y = Br1[48];

    acc0 = __builtin_amdgcn_wmma_f32_16x16x4_f32(false, a, false, b0, (short)0, acc0, false, false);
    acc1 = __builtin_amdgcn_wmma_f32_16x16x4_f32(false, a, false, b1, (short)0, acc1, false, false);
    acc2 = __builtin_amdgcn_wmma_f32_16x16x4_f32(false, a, false, b2, (short)0, acc2, false, false);
    acc3 = __builtin_amdgcn_wmma_f32_16x16x4_f32(false, a, false, b3, (short)0, acc3, false, false);
  }

  const int crow = tileM + 8 * half;
#pragma unroll
  for (int r = 0; r < 8; ++r) {
    size_t base = (size_t)(crow + r) * Nc + tileN + l16;
    C[base]      = acc0[r];
    C[base + 16] = acc1[r];
    C[base + 32] = acc2[r];
    C[base + 48] = acc3[r];
  }
}

// =====================================================================
// Init: mkey = key(-inf), denom = 0, agg = 0 (grid-stride over agg)
// =====================================================================
__global__ void init_state(unsigned* __restrict__ mkey, float* __restrict__ denom,
                           float* __restrict__ agg, int nmk, size_t nagg) {
  size_t i = (size_t)blockIdx.x * blockDim.x + threadIdx.x;
  if (i < (size_t)nmk) { mkey[i] = NEG_INF_KEY; denom[i] = 0.0f; }
  size_t stride = (size_t)gridDim.x * blockDim.x;
  for (size_t j = i; j < nagg; j += stride) agg[j] = 0.0f;
}

// =====================================================================
// Per-edge attention logits + segment max (one wave per edge, lanes over C)
// logit[e,h] = att[h] . leaky_relu(xl[src] + xr[dst])
// =====================================================================
__global__ __launch_bounds__(256)
void edge_logits(const long long* __restrict__ ei, int E, int ET,
                 const float* __restrict__ xl, const float* __restrict__ xr,
                 const float* __restrict__ att, int C,
                 float* __restrict__ logits, unsigned* __restrict__ mkey) {
  int e = blockIdx.x * 8 + (threadIdx.x >> 5);
  if (e >= ET) return;
  const int lane = threadIdx.x & 31;
  const int src = (e < E) ? (int)ei[e]     : (e - E);
  const int dst = (e < E) ? (int)ei[E + e] : (e - E);

#pragma unroll
  for (int h = 0; h < HEADS; ++h) {
    const float* __restrict__ pl = xl + ((size_t)src * HEADS + h) * C;
    const float* __restrict__ pr = xr + ((size_t)dst * HEADS + h) * C;
    const float* __restrict__ pa = att + (size_t)h * C;
    float s = 0.0f;
    for (int c = lane; c < C; c += 32) {
      float v = pl[c] + pr[c];
      v = (v > 0.0f) ? v : NEG_SLOPE * v;
      s += pa[c] * v;
    }
#pragma unroll
    for (int off = 16; off; off >>= 1) s += __shfl_xor(s, off, 32);
    if (lane == 0) {
      logits[(size_t)e * HEADS + h] = s;
      atomicMax(&mkey[(size_t)dst * HEADS + h], fkey(s));
    }
  }
}

// =====================================================================
// Per-edge: ea = exp(logit - max[dst]); denom[dst] += ea;
//           agg[dst,h,:] += ea * xl[src,h,:]   (alpha division deferred)
// =====================================================================
__global__ __launch_bounds__(256)
void edge_accum(const long long* __restrict__ ei, int E, int ET,
                const float* __restrict__ xl, const float* __restrict__ logits,
                const unsigned* __restrict__ mkey, float* __restrict__ denom,
                float* __restrict__ agg, int C) {
  int e = blockIdx.x * 8 + (threadIdx.x >> 5);
  if (e >= ET) return;
  const int lane = threadIdx.x & 31;
  const int src = (e < E) ? (int)ei[e]     : (e - E);
  const int dst = (e < E) ? (int)ei[E + e] : (e - E);

  float ea_l = 0.0f;
  if (lane < HEADS) {
    float lg = logits[(size_t)e * HEADS + lane];
    float mv = funkey(mkey[(size_t)dst * HEADS + lane]);
    ea_l = __expf(lg - mv);
    atomicAdd(&denom[(size_t)dst * HEADS + lane], ea_l);
  }
  float ea[HEADS];
#pragma unroll
  for (int h = 0; h < HEADS; ++h) ea[h] = __shfl(ea_l, h, 32);

#pragma unroll
  for (int h = 0; h < HEADS; ++h) {
    const float* __restrict__ pl = xl + ((size_t)src * HEADS + h) * C;
    float* __restrict__ pd = agg + ((size_t)dst * HEADS + h) * C;
    for (int c = lane; c < C; c += 32)
      atomicAdd(&pd[c], ea[h] * pl[c]);
  }
}

// =====================================================================
// Finalize: out[n,c] = act( mean_h(agg[n,h,c]/denom[n,h]) + bias[c] )
// act: 0 = leaky_relu(0.2), 1 = tanh
// =====================================================================
__global__ void finalize_nodes(const float* __restrict__ agg, const float* __restrict__ denom,
                               const float* __restrict__ bias, float* __restrict__ out,
                               int Nn, int C, int act) {
  int idx = blockIdx.x * blockDim.x + threadIdx.x;
  if (idx >= Nn * C) return;
  int n = idx / C, c = idx - n * C;
  float s = 0.0f;
#pragma unroll
  for (int h = 0; h < HEADS; ++h)
    s += agg[((size_t)n * HEADS + h) * C + c] / denom[(size_t)n * HEADS + h];
  s = s * (1.0f / HEADS) + bias[c];
  s = (act == 0) ? ((s > 0.0f) ? s : NEG_SLOPE * s) : tanhf(s);
  out[idx] = s;
}

// =====================================================================
extern "C" void kernel_launch(void* const* d_in, const int* in_sizes, int n_in,
                              void* d_out, int out_size, void* d_ws, size_t ws_size,
                              hipStream_t stream) {
  const float*     x    = (const float*)d_in[0];
  const long long* ei   = (const long long*)d_in[1];
  const float*     Wl1  = (const float*)d_in[2];
  const float*     Wr1  = (const float*)d_in[3];
  const float*     att1 = (const float*)d_in[4];
  const float*     b1   = (const float*)d_in[5];
  const float*     Wl2  = (const float*)d_in[6];
  const float*     Wr2  = (const float*)d_in[7];
  const float*     att2 = (const float*)d_in[8];
  const float*     b2   = (const float*)d_in[9];

  const int IN  = 512, HID = 256, OUT = 128;
  const int N   = in_sizes[0] / IN;       // 10000
  const int E   = in_sizes[1] / 2;        // 160000
  const int ET  = E + N;                  // + self loops

  // ---- workspace carve-out (all fp32 unless noted) ----
  char* w = (char*)d_ws;
  size_t off = 0;
  auto alloc = [&](size_t bytes) -> void* {
    void* p = w + off;
    off += (bytes + 255) & ~(size_t)255;
    return p;
  };
  float*    xl     = (float*)alloc((size_t)N * HEADS * HID * 4);  // max-size, reused L2
  float*    xr     = (float*)alloc((size_t)N * HEADS * HID * 4);
  float*    agg    = (float*)alloc((size_t)N * HEADS * HID * 4);
  float*    logits = (float*)alloc((size_t)ET * HEADS * 4);
  unsigned* mkey   = (unsigned*)alloc((size_t)N * HEADS * 4);
  float*    denom  = (float*)alloc((size_t)N * HEADS * 4);
  float*    h1     = (float*)alloc((size_t)N * HID * 4);
  (void)ws_size; (void)n_in; (void)out_size;

  const dim3 blk256(256);
  const int  mtiles    = N / 16;                       // N % 16 == 0
  const int  gy        = (mtiles + 7) / 8;
  const dim3 gemm_g1(HEADS * HID / 64, gy);            // Nc = 1024
  const dim3 gemm_g2(HEADS * OUT / 64, gy);            // Nc = 512
  const int  edge_blocks = (ET + 7) / 8;

  // ================= Layer 1: GATv2(IN=512 -> HID=256, H=4) =================
  gemm_f32_wmma<<<gemm_g1, blk256, 0, stream>>>(x, Wl1, xl, N, IN, HEADS * HID);
  gemm_f32_wmma<<<gemm_g1, blk256, 0, stream>>>(x, Wr1, xr, N, IN, HEADS * HID);
  init_state<<<4096, blk256, 0, stream>>>(mkey, denom, agg, N * HEADS, (size_t)N * HEADS * HID);
  edge_logits<<<edge_blocks, blk256, 0, stream>>>(ei, E, ET, xl, xr, att1, HID, logits, mkey);
  edge_accum<<<edge_blocks, blk256, 0, stream>>>(ei, E, ET, xl, logits, mkey, denom, agg, HID);
  finalize_nodes<<<(N * HID + 255) / 256, blk256, 0, stream>>>(agg, denom, b1, h1, N, HID, /*leaky*/0);

  // ================= Layer 2: GATv2(HID=256 -> OUT=128, H=4) =================
  gemm_f32_wmma<<<gemm_g2, blk256, 0, stream>>>(h1, Wl2, xl, N, HID, HEADS * OUT);
  gemm_f32_wmma<<<gemm_g2, blk256, 0, stream>>>(h1, Wr2, xr, N, HID, HEADS * OUT);
  init_state<<<4096, blk256, 0, stream>>>(mkey, denom, agg, N * HEADS, (size_t)N * HEADS * HID);
  edge_logits<<<edge_blocks, blk256, 0, stream>>>(ei, E, ET, xl, xr, att2, OUT, logits, mkey);
  edge_accum<<<edge_blocks, blk256, 0, stream>>>(ei, E, ET, xl, logits, mkey, denom, agg, OUT);
  finalize_nodes<<<(N * OUT + 255) / 256, blk256, 0, stream>>>(agg, denom, b2, (float*)d_out, N, OUT, /*tanh*/1);
}